// AGSG_20452634264233
// MI455X (gfx1250) — compile-verified
//
#include <hip/hip_runtime.h>

// Problem constants (from reference)
#define NN 512   // NUM_NODES
#define CC 64    // CHANNELS
#define BB 16    // BATCH
#define TT 12    // T

typedef __attribute__((ext_vector_type(2))) float v2f;
typedef __attribute__((ext_vector_type(8))) float v8f;

// D = A(16x4) * B(4x16) + C, fp32 in/out (V_WMMA_F32_16X16X4_F32)
__device__ __forceinline__ v8f wmma4(v2f a, v2f b, v8f c) {
  return __builtin_amdgcn_wmma_f32_16x16x4_f32(
      /*neg_a=*/false, a, /*neg_b=*/false, b,
      /*c_mod=*/(short)0, c, /*reuse_a=*/false, /*reuse_b=*/false);
}

// ---------------------------------------------------------------------------
// Gram: S[n,m] = sum_c M[c,n] * M[c,m].  M is [C, N] row-major (n contiguous).
// One wave per 16x16 tile of S. A' = M^T so A'[n,c] = M[c,n].
// Per ISA layout: lane l (li=l&15, half=l>>4): a = {A'[row0+li, kk], A'[row0+li, kk+1]},
// b = {B[kk, col0+li], B[kk+1, col0+li]}, kk = k + 2*half.
// ---------------------------------------------------------------------------
__global__ void gram_kernel(const float* __restrict__ M, float* __restrict__ S) {
  const int lane = threadIdx.x;
  const int half = lane >> 4;
  const int li   = lane & 15;
  const int ti   = blockIdx.y * blockDim.y + threadIdx.y;  // 0..31
  const int tj   = blockIdx.x;                             // 0..31
  const int row0 = ti * 16, col0 = tj * 16;
  v8f acc = {};
#pragma unroll 4
  for (int k = 0; k < CC; k += 4) {
    const int kk = k + 2 * half;
    v2f a, b;
    a.x = M[(kk    ) * NN + row0 + li];   // A'[row, kk]   = M[kk, row]
    a.y = M[(kk + 1) * NN + row0 + li];
    b.x = M[(kk    ) * NN + col0 + li];   // B[kk, col]
    b.y = M[(kk + 1) * NN + col0 + li];
    acc = wmma4(a, b, acc);
  }
#pragma unroll
  for (int i = 0; i < 8; ++i)
    S[(size_t)(row0 + i + 8 * half) * NN + col0 + li] = acc[i];
}

// ---------------------------------------------------------------------------
// Row kernel: S_w = softmax(relu(S) with diag forced to 0.1), write:
//   P = S_w          (A^1)
//   Tm = I + S_w     (f(2) = sum_{k<2} A^k)
// One 256-thread block per row, 2 elements per thread.
// ---------------------------------------------------------------------------
__global__ void rowsoft_kernel(const float* __restrict__ S,
                               float* __restrict__ P, float* __restrict__ Tm) {
  __shared__ float red[256];
  const int n = blockIdx.x, tid = threadIdx.x;
  float v0 = fmaxf(S[(size_t)n * NN + tid], 0.f);
  float v1 = fmaxf(S[(size_t)n * NN + tid + 256], 0.f);
  if (tid == n)       v0 = 0.1f;
  if (tid + 256 == n) v1 = 0.1f;
  red[tid] = fmaxf(v0, v1); __syncthreads();
  for (int s = 128; s > 0; s >>= 1) {
    if (tid < s) red[tid] = fmaxf(red[tid], red[tid + s]);
    __syncthreads();
  }
  const float m = red[0]; __syncthreads();
  const float e0 = __expf(v0 - m), e1 = __expf(v1 - m);
  red[tid] = e0 + e1; __syncthreads();
  for (int s = 128; s > 0; s >>= 1) {
    if (tid < s) red[tid] += red[tid + s];
    __syncthreads();
  }
  const float inv = 1.f / red[0];
  const float w0 = e0 * inv, w1 = e1 * inv;
  P[(size_t)n * NN + tid]        = w0;
  P[(size_t)n * NN + tid + 256]  = w1;
  Tm[(size_t)n * NN + tid]       = w0 + ((tid == n)       ? 1.f : 0.f);
  Tm[(size_t)n * NN + tid + 256] = w1 + ((tid + 256 == n) ? 1.f : 0.f);
}

// ---------------------------------------------------------------------------
// D = A @ B (+ Cinit).  512x512x512 fp32 via WMMA, one wave per 16x16 tile.
// A, B row-major [N, N].
// ---------------------------------------------------------------------------
__global__ void matmul_kernel(const float* __restrict__ A, const float* __restrict__ Bm,
                              const float* __restrict__ Cinit, float* __restrict__ D) {
  const int lane = threadIdx.x;
  const int half = lane >> 4;
  const int li   = lane & 15;
  const int ti   = blockIdx.y * blockDim.y + threadIdx.y;  // 0..31
  const int tj   = blockIdx.x;                             // 0..31
  const int row0 = ti * 16, col0 = tj * 16;
  v8f acc;
  if (Cinit) {
#pragma unroll
    for (int i = 0; i < 8; ++i)
      acc[i] = Cinit[(size_t)(row0 + i + 8 * half) * NN + col0 + li];
  } else {
    acc = (v8f){};
  }
  const float* ap = A + (size_t)(row0 + li) * NN;  // this lane's A row
#pragma unroll 4
  for (int k = 0; k < NN; k += 4) {
    const int kk = k + 2 * half;
    v2f a, b;
    a.x = ap[kk];                         // contiguous pair -> b64 load
    a.y = ap[kk + 1];
    b.x = Bm[(size_t)kk * NN + col0 + li];
    b.y = Bm[(size_t)(kk + 1) * NN + col0 + li];
    acc = wmma4(a, b, acc);
  }
#pragma unroll
  for (int i = 0; i < 8; ++i)
    D[(size_t)(row0 + i + 8 * half) * NN + col0 + li] = acc[i];
}

// ---------------------------------------------------------------------------
// xs[b,n] = sum_{c,t} x[b,c,n,t].   x flat: ((b*C + c)*N + n)*T + t
// ---------------------------------------------------------------------------
__global__ void xsum_kernel(const float* __restrict__ x, float* __restrict__ xs) {
  const int idx = blockIdx.x * blockDim.x + threadIdx.x;  // b*N + n
  if (idx >= BB * NN) return;
  const int b = idx >> 9, n = idx & (NN - 1);
  float s = 0.f;
  const float* base = x + ((size_t)b * CC * NN + n) * TT;
  for (int c = 0; c < CC; ++c) {
    const float* p = base + (size_t)c * NN * TT;
#pragma unroll
    for (int t = 0; t < TT; ++t) s += p[t];
  }
  xs[idx] = s;
}

// ---------------------------------------------------------------------------
// out[b,n,:] = softmax( relu( xs[b,n]/8 * (Tm[n,:] + P[n,:]) ) )
// G = Tm + P  (= sum_{k=0..512} S_w^k), fused here. One block per (b,n) row.
// ---------------------------------------------------------------------------
__global__ void final_kernel(const float* __restrict__ Tm, const float* __restrict__ P,
                             const float* __restrict__ xs, float* __restrict__ out) {
  __shared__ float red[256];
  const int bn = blockIdx.x;          // 0 .. B*N-1
  const int n  = bn & (NN - 1);
  const int tid = threadIdx.x;
  const float scale = xs[bn] * 0.125f;   // 1/sqrt(64)
  const float g0 = Tm[(size_t)n * NN + tid]       + P[(size_t)n * NN + tid];
  const float g1 = Tm[(size_t)n * NN + tid + 256] + P[(size_t)n * NN + tid + 256];
  const float v0 = fmaxf(scale * g0, 0.f);
  const float v1 = fmaxf(scale * g1, 0.f);
  red[tid] = fmaxf(v0, v1); __syncthreads();
  for (int s = 128; s > 0; s >>= 1) {
    if (tid < s) red[tid] = fmaxf(red[tid], red[tid + s]);
    __syncthreads();
  }
  const float m = red[0]; __syncthreads();
  const float e0 = __expf(v0 - m), e1 = __expf(v1 - m);
  red[tid] = e0 + e1; __syncthreads();
  for (int s = 128; s > 0; s >>= 1) {
    if (tid < s) red[tid] += red[tid + s];
    __syncthreads();
  }
  const float inv = 1.f / red[0];
  float* orow = out + (size_t)bn * NN;
  orow[tid]       = e0 * inv;
  orow[tid + 256] = e1 * inv;
}

extern "C" void kernel_launch(void* const* d_in, const int* in_sizes, int n_in,
                              void* d_out, int out_size, void* d_ws, size_t ws_size,
                              hipStream_t stream) {
  const float* x   = (const float*)d_in[0];   // [16,64,512,12]
  const float* mem = (const float*)d_in[1];   // [64,512]
  float* out = (float*)d_out;
  float* ws  = (float*)d_ws;

  const size_t MAT = (size_t)NN * NN;         // 262144 floats
  float* g_S  = ws;                           // raw Gram; reused as P ping
  float* g_P0 = ws + MAT;
  float* g_T0 = ws + 2 * MAT;
  float* g_T1 = ws + 3 * MAT;
  float* g_xs = ws + 4 * MAT;                 // 8192 floats

  const dim3 mblock(32, 4);   // 4 waves/block, one wave per 16x16 tile
  const dim3 mgrid(32, 8);    // 32x32 tiles

  // 1) Gram + row softmax  ->  P = S_w,  T = I + S_w  (= f(2))
  gram_kernel<<<mgrid, mblock, 0, stream>>>(mem, g_S);
  rowsoft_kernel<<<NN, 256, 0, stream>>>(g_S, g_P0, g_T0);

  // 2) Geometric-series doubling: invariant T = sum_{k<m} A^k, P = A^m.
  float* Pcur = g_P0; float* Pnext = g_S;
  float* Tcur = g_T0; float* Tnext = g_T1;

  // m = 2: need P = A^2 (T already = f(2))
  matmul_kernel<<<mgrid, mblock, 0, stream>>>(Pcur, Pcur, nullptr, Pnext);
  { float* t = Pcur; Pcur = Pnext; Pnext = t; }

  // 8 doublings: m = 2 -> 512
  for (int step = 0; step < 8; ++step) {
    matmul_kernel<<<mgrid, mblock, 0, stream>>>(Pcur, Tcur, Tcur, Tnext); // T' = T + P@T
    matmul_kernel<<<mgrid, mblock, 0, stream>>>(Pcur, Pcur, nullptr, Pnext); // P' = P@P
    float* t;
    t = Tcur; Tcur = Tnext; Tnext = t;
    t = Pcur; Pcur = Pnext; Pnext = t;
  }
  // Now Tcur = sum_{k=0..511} A^k, Pcur = A^512; G = Tcur + Pcur (fused into final).

  // 3) xs[b,n] and fused scale/relu/softmax output
  xsum_kernel<<<(BB * NN) / 256, 256, 0, stream>>>(x, g_xs);
  final_kernel<<<BB * NN, 256, 0, stream>>>(Tcur, Pcur, g_xs, out);
}